// ClsEncoder_80960133530358
// MI455X (gfx1250) — compile-verified
//
#include <hip/hip_runtime.h>
#include <hip/hip_bf16.h>

typedef __attribute__((ext_vector_type(2))) float v2f;
typedef __attribute__((ext_vector_type(8))) float v8f;

#define GEMM_WAVES 4      // waves per block, one 16-row M-tile each
#define PAIR_STRIDE 160   // floats per K-pair row in LDS (64*2 data + 32 pad
                          // -> lane halves hit disjoint bank sets {2n},{2n+32})

// ---------------------------------------------------------------------------
// Degree count: deg[dst[e]] += 1  (deg buffer pre-zeroed via memset)
// ---------------------------------------------------------------------------
__global__ void deg_kernel(const long long* __restrict__ dst,
                           float* __restrict__ deg, int E) {
    for (int e = blockIdx.x * blockDim.x + threadIdx.x; e < E;
         e += gridDim.x * blockDim.x) {
        __builtin_prefetch(&dst[e + 8192], 0, 0);   // global_prefetch_b8
        atomicAdd(&deg[(int)dst[e]], 1.0f);
    }
}

// dinv[i] = rsqrt(deg[i] + 1)   (in place)
__global__ void dinv_kernel(float* __restrict__ d, int N) {
    for (int i = blockIdx.x * blockDim.x + threadIdx.x; i < N;
         i += gridDim.x * blockDim.x) {
        d[i] = rsqrtf(d[i] + 1.0f);
    }
}

// ---------------------------------------------------------------------------
// H[N x 64] = X[N x K] @ W[K x 64] via V_WMMA_F32_16X16X4_F32 (full f32).
// Block = 128 threads = 4 waves; W staged in LDS *pre-paired* so every B
// operand is one aligned ds_load_b64 straight into an even VGPR pair:
//   sWp[(k>>1)*PAIR_STRIDE + n*2 + (k&1)] = W[k][n]
// A 16x4 layout: lane half h=(lane>>4): VGPR0=K(2h), VGPR1=K(2h+1), M=lane&15.
// B 4x16 mirrors A with N=lane&15.  D: VGPR r = row (8h + r), col = lane&15.
// ---------------------------------------------------------------------------
__global__ void __launch_bounds__(128)
gemm64_wmma(const float* __restrict__ X, const float* __restrict__ W,
            float* __restrict__ H, int N, int K) {
    __shared__ float sWp[64 * PAIR_STRIDE];      // up to K=128 -> 64 pair-rows
    const int tid = threadIdx.x;
    for (int i = tid; i < K * 64; i += blockDim.x) {
        const int k = i >> 6;
        const int n = i & 63;
        sWp[(k >> 1) * PAIR_STRIDE + n * 2 + (k & 1)] = W[i];
    }
    __syncthreads();

    const int wave = tid >> 5;
    const int lane = tid & 31;
    const int mtiles = (N + 15) >> 4;
    const int mt = blockIdx.x * GEMM_WAVES + wave;
    if (mt >= mtiles) return;                    // whole-wave uniform branch

    const int m  = lane & 15;                    // row-in-tile (A), col (B/D)
    const int kb = (lane >> 4) << 1;             // K sub-offset for lane half

    long long arow = (long long)mt * 16 + m;
    if (arow >= N) arow = N - 1;                 // clamp; clamped rows not stored
    const float* __restrict__ xrow = X + arow * (long long)K;

    v8f acc0 = {}, acc1 = {}, acc2 = {}, acc3 = {};
    for (int k0 = 0; k0 < K; k0 += 4) {
        const v2f a = *(const v2f*)&xrow[k0 + kb];           // 8B aligned
        const float* __restrict__ bp =
            &sWp[((k0 + kb) >> 1) * PAIR_STRIDE + m * 2];
        const v2f b0 = *(const v2f*)(bp + 0);
        const v2f b1 = *(const v2f*)(bp + 32);
        const v2f b2 = *(const v2f*)(bp + 64);
        const v2f b3 = *(const v2f*)(bp + 96);
        acc0 = __builtin_amdgcn_wmma_f32_16x16x4_f32(false, a, false, b0,
                                                     (short)0, acc0, false, false);
        acc1 = __builtin_amdgcn_wmma_f32_16x16x4_f32(false, a, false, b1,
                                                     (short)0, acc1, false, false);
        acc2 = __builtin_amdgcn_wmma_f32_16x16x4_f32(false, a, false, b2,
                                                     (short)0, acc2, false, false);
        acc3 = __builtin_amdgcn_wmma_f32_16x16x4_f32(false, a, false, b3,
                                                     (short)0, acc3, false, false);
    }

    const int rbase = mt * 16 + ((lane >> 4) << 3);
    const int col = lane & 15;
#pragma unroll
    for (int r = 0; r < 8; ++r) {
        const int row = rbase + r;
        if (row < N) {
            float* __restrict__ hp = H + (long long)row * 64 + col;
            hp[0]  = acc0[r];
            hp[16] = acc1[r];
            hp[32] = acc2[r];
            hp[48] = acc3[r];
        }
    }
}

// ---------------------------------------------------------------------------
// Edge scatter: AGG[dst] += H[src] * dinv[src]*dinv[dst]
// One thread per (edge, float4 quad): 16 consecutive lanes cover one full
// 256B H row (coalesced gather). AGG fits in the 192MB L2, so the f32
// atomics resolve at L2 atomic units.
// ---------------------------------------------------------------------------
__global__ void scatter_kernel(const long long* __restrict__ src,
                               const long long* __restrict__ dst,
                               const float* __restrict__ dinv,
                               const float* __restrict__ H,
                               float* __restrict__ AGG, int E) {
    const long long total = (long long)E * 16;
    for (long long t = (long long)blockIdx.x * blockDim.x + threadIdx.x;
         t < total; t += (long long)gridDim.x * blockDim.x) {
        const int e = (int)(t >> 4);
        const int q = ((int)t & 15) << 2;
        __builtin_prefetch(&src[e + 4096], 0, 0);
        __builtin_prefetch(&dst[e + 4096], 0, 0);
        const int s = (int)src[e];
        const int d = (int)dst[e];
        const float norm = dinv[s] * dinv[d];
        const float4 hv = *(const float4*)(H + (long long)s * 64 + q);
        float* __restrict__ ap = AGG + (long long)d * 64 + q;
        atomicAdd(ap + 0, hv.x * norm);
        atomicAdd(ap + 1, hv.y * norm);
        atomicAdd(ap + 2, hv.z * norm);
        atomicAdd(ap + 3, hv.w * norm);
    }
}

// AGG = (relu?)(AGG + H*dinv^2 + bias), in place; float4 per thread.
__global__ void finalize_kernel(float* __restrict__ AGG,
                                const float* __restrict__ H,
                                const float* __restrict__ dinv,
                                const float* __restrict__ bias,
                                int N, int do_relu) {
    const long long total = (long long)N * 16;          // float4 quads
    for (long long t = (long long)blockIdx.x * blockDim.x + threadIdx.x;
         t < total; t += (long long)gridDim.x * blockDim.x) {
        const int node = (int)(t >> 4);
        const int q = ((int)t & 15) << 2;
        const float di = dinv[node];
        const float di2 = di * di;
        const long long idx = (long long)node * 64 + q;
        const float4 av = *(const float4*)(AGG + idx);
        const float4 hv = *(const float4*)(H + idx);
        const float4 bv = *(const float4*)(bias + q);
        float4 v;
        v.x = av.x + hv.x * di2 + bv.x;
        v.y = av.y + hv.y * di2 + bv.y;
        v.z = av.z + hv.z * di2 + bv.z;
        v.w = av.w + hv.w * di2 + bv.w;
        if (do_relu) {
            v.x = fmaxf(v.x, 0.0f); v.y = fmaxf(v.y, 0.0f);
            v.z = fmaxf(v.z, 0.0f); v.w = fmaxf(v.w, 0.0f);
        }
        *(float4*)(AGG + idx) = v;
    }
}

// Stage 1 of mean: partial column sums of (AGG + H*dinv^2); 64 threads/block.
__global__ void colsum_partial(const float* __restrict__ AGG,
                               const float* __restrict__ H,
                               const float* __restrict__ dinv,
                               float* __restrict__ part, int N) {
    const int f = threadIdx.x;                 // 0..63
    float acc = 0.0f;
    for (int i = blockIdx.x; i < N; i += gridDim.x) {
        const float di = dinv[i];
        const long long idx = (long long)i * 64 + f;
        acc += AGG[idx] + H[idx] * di * di;
    }
    part[blockIdx.x * 64 + f] = acc;
}

// Stage 2 of mean: out[f] = sum(part)/N + b2[f]
__global__ void colsum_final(const float* __restrict__ part,
                             const float* __restrict__ b2,
                             float* __restrict__ out, int nb, float invN) {
    const int f = threadIdx.x;                 // 0..63
    float acc = 0.0f;
    for (int b = 0; b < nb; ++b) acc += part[b * 64 + f];
    out[f] = acc * invN + b2[f];
}

// ---------------------------------------------------------------------------
extern "C" void kernel_launch(void* const* d_in, const int* in_sizes, int n_in,
                              void* d_out, int out_size, void* d_ws, size_t ws_size,
                              hipStream_t stream) {
    const float*     X     = (const float*)d_in[0];       // [N,128] f32
    const long long* edges = (const long long*)d_in[1];   // [2,E]  i64
    const float*     W1    = (const float*)d_in[2];       // [128,64]
    const float*     b1    = (const float*)d_in[3];       // [64]
    const float*     W2    = (const float*)d_in[4];       // [64,64]
    const float*     b2    = (const float*)d_in[5];       // [64]
    float*           out   = (float*)d_out;               // [64]

    const int N = in_sizes[0] / 128;
    const int E = in_sizes[1] / 2;
    const long long* src = edges;
    const long long* dst = edges + E;

    // Workspace layout (floats): dinv[N] | H[N*64] | AGG[N*64] | part[1024*64]
    float* ws   = (float*)d_ws;
    float* dinv = ws;
    size_t offH = ((size_t)N + 3) & ~(size_t)3;
    float* H    = ws + offH;
    float* AGG  = H + (size_t)N * 64;
    float* part = AGG + (size_t)N * 64;
    const int RB = 1024;

    // Degrees -> dinv
    hipMemsetAsync(dinv, 0, (size_t)N * sizeof(float), stream);
    deg_kernel<<<2048, 256, 0, stream>>>(dst, dinv, E);
    dinv_kernel<<<512, 256, 0, stream>>>(dinv, N);

    const int mtiles  = (N + 15) / 16;
    const int gblocks = (mtiles + GEMM_WAVES - 1) / GEMM_WAVES;

    // ---- Layer 1 ----
    gemm64_wmma<<<gblocks, 128, 0, stream>>>(X, W1, H, N, 128);
    hipMemsetAsync(AGG, 0, (size_t)N * 64 * sizeof(float), stream);
    scatter_kernel<<<8192, 256, 0, stream>>>(src, dst, dinv, H, AGG, E);
    finalize_kernel<<<2048, 256, 0, stream>>>(AGG, H, dinv, b1, N, 1); // AGG=h1

    // ---- Layer 2 ----
    gemm64_wmma<<<gblocks, 128, 0, stream>>>(AGG, W2, H, N, 64);       // H=h1@W2
    hipMemsetAsync(AGG, 0, (size_t)N * 64 * sizeof(float), stream);
    scatter_kernel<<<8192, 256, 0, stream>>>(src, dst, dinv, H, AGG, E);

    // Mean over nodes, folding self-loop term + b2
    colsum_partial<<<RB, 64, 0, stream>>>(AGG, H, dinv, part, N);
    colsum_final<<<1, 64, 0, stream>>>(part, b2, out, RB, 1.0f / (float)N);
}